// GCNlayer_6665789243397
// MI455X (gfx1250) — compile-verified
//
#include <hip/hip_runtime.h>

#define N_NODES 100000
#define N_EDGES 1600000
#define D 128
#define N_TILES (N_NODES / 16)   // 6250, exact
#define WPITCH 132               // LDS row pitch (floats): conflict-free b64 reads
#define MLP_WAVES 16             // waves per MLP block (512 threads)

typedef __attribute__((ext_vector_type(2))) float v2f;
typedef __attribute__((ext_vector_type(8))) float v8f;

// Hardware no-return f32 atomic add (VGLOBAL GV mode, STOREcnt-tracked;
// implicit wait-idle at s_endpgm covers completion).
__device__ __forceinline__ void hw_atomic_add_f32(float* p, float v) {
    asm volatile("global_atomic_add_f32 %0, %1, off"
                 :
                 : "v"(p), "v"(v)
                 : "memory");
}

// ---------------- degree: deg[dst] += 1 ----------------
__global__ void deg_kernel(const int* __restrict__ dst, float* __restrict__ deg) {
    int e = blockIdx.x * blockDim.x + threadIdx.x;
    if (e < N_EDGES) {
        hw_atomic_add_f32(&deg[dst[e]], 1.0f);
    }
}

// ---------------- dis = deg > 0 ? rsqrt(deg) : 0 ----------------
__global__ void dis_kernel(const float* __restrict__ deg, float* __restrict__ dis) {
    int i = blockIdx.x * blockDim.x + threadIdx.x;
    if (i < N_NODES) {
        float d = deg[i];
        dis[i] = (d > 0.0f) ? rsqrtf(d) : 0.0f;
    }
}

// ---------------- scatter: agg[dst] += nh[src] * dis[src] ----------------
// one 32-lane wave per edge; each lane handles 4 features (stride 32 -> four
// coalesced 128B accesses per wave, both for the gather and the atomics)
__global__ void scatter_kernel(const float* __restrict__ nh,
                               const int* __restrict__ src,
                               const int* __restrict__ dst,
                               const float* __restrict__ dis,
                               float* __restrict__ agg) {
    long long gid = (long long)blockIdx.x * blockDim.x + threadIdx.x;
    int e    = (int)(gid >> 5);
    int lane = (int)(gid & 31);
    if (e >= N_EDGES) return;           // wave-uniform branch
    int s = src[e];
    int d = dst[e];
    float w = dis[s];
    const float* np = nh  + (long long)s * D;
    float*       ap = agg + (long long)d * D;
#pragma unroll
    for (int j = 0; j < 4; ++j) {
        int f = lane + 32 * j;
        hw_atomic_add_f32(&ap[f], np[f] * w);
    }
}

// ---------------- fused scale + 2-layer MLP, fp32 WMMA ----------------
// io holds agg on entry; n_h written in place (each wave consumes its 16 rows
// fully before storing).  One wave -> 16 rows x 128 cols via 8 N-tiles of
// V_WMMA_F32_16X16X4_F32, K-loop step 4.  16 waves/block (4 per SIMD32).
__global__ __launch_bounds__(32 * MLP_WAVES)
void mlp_kernel(const float* __restrict__ dis,
                const float* __restrict__ W1,
                const float* __restrict__ b1,
                const float* __restrict__ W2,
                const float* __restrict__ b2,
                float* __restrict__ io) {
    extern __shared__ float smem[];
    float* Wt1  = smem;                      // [128][WPITCH]  W1 transposed: Wt1[n][k]
    float* Wt2  = Wt1 + 128 * WPITCH;        // [128][WPITCH]
    float* b1s  = Wt2 + 128 * WPITCH;        // [128]
    float* b2s  = b1s + 128;                 // [128]
    float* hall = b2s + 128;                 // MLP_WAVES * [16][WPITCH]

    int tid = threadIdx.x;

    // cooperative load of weights (transposed) and biases into LDS
    for (int idx = tid; idx < 128 * 128; idx += 32 * MLP_WAVES) {
        int k = idx >> 7;
        int n = idx & 127;
        Wt1[n * WPITCH + k] = W1[idx];
        Wt2[n * WPITCH + k] = W2[idx];
    }
    if (tid < 128) { b1s[tid] = b1[tid]; b2s[tid] = b2[tid]; }
    __syncthreads();

    int wave = tid >> 5;
    int lane = tid & 31;
    int tile = blockIdx.x * MLP_WAVES + wave;
    if (tile >= N_TILES) return;

    float* h = hall + wave * (16 * WPITCH);

    int mn    = lane & 15;              // A: row M; B/C/D: col N
    int kb    = (lane < 16) ? 0 : 2;    // K sub-offset for A/B halves
    int mbase = (lane < 16) ? 0 : 8;    // C/D row base
    int row0  = tile * 16;

    float scale = dis[row0 + mn];       // dis of this lane's A-row
    const float* arow = io + (long long)(row0 + mn) * D;

    v8f zero = {0.f, 0.f, 0.f, 0.f, 0.f, 0.f, 0.f, 0.f};
    v8f acc[8];
#pragma unroll
    for (int t = 0; t < 8; ++t) acc[t] = zero;

    // ---- layer 1: h = relu((agg*dis) @ W1 + b1) ----
    for (int k0 = 0; k0 < 128; k0 += 4) {
        v2f a;
        a.x = arow[k0 + kb]     * scale;
        a.y = arow[k0 + kb + 1] * scale;
#pragma unroll
        for (int nt = 0; nt < 8; ++nt) {
            v2f b;
            const float* wr = Wt1 + (nt * 16 + mn) * WPITCH + k0 + kb;
            b.x = wr[0];
            b.y = wr[1];
            acc[nt] = __builtin_amdgcn_wmma_f32_16x16x4_f32(
                false, a, false, b, (short)0, acc[nt], false, false);
        }
    }

    // bias + relu, stash hidden tile in per-wave LDS (A-operand layout next)
#pragma unroll
    for (int nt = 0; nt < 8; ++nt) {
        float bb = b1s[nt * 16 + mn];
#pragma unroll
        for (int r = 0; r < 8; ++r) {
            float v = acc[nt][r] + bb;
            v = (v > 0.0f) ? v : 0.0f;
            h[(mbase + r) * WPITCH + nt * 16 + mn] = v;
        }
    }

    // ---- layer 2: n_h = h @ W2 + b2 ----
    v8f acc2[8];
#pragma unroll
    for (int t = 0; t < 8; ++t) acc2[t] = zero;

    for (int k0 = 0; k0 < 128; k0 += 4) {
        v2f a;
        const float* hr = h + mn * WPITCH + k0 + kb;
        a.x = hr[0];
        a.y = hr[1];
#pragma unroll
        for (int nt = 0; nt < 8; ++nt) {
            v2f b;
            const float* wr = Wt2 + (nt * 16 + mn) * WPITCH + k0 + kb;
            b.x = wr[0];
            b.y = wr[1];
            acc2[nt] = __builtin_amdgcn_wmma_f32_16x16x4_f32(
                false, a, false, b, (short)0, acc2[nt], false, false);
        }
    }

    // store n_h in place over agg rows of this tile
#pragma unroll
    for (int nt = 0; nt < 8; ++nt) {
        float bb = b2s[nt * 16 + mn];
#pragma unroll
        for (int r = 0; r < 8; ++r) {
            io[(long long)(row0 + mbase + r) * D + nt * 16 + mn] = acc2[nt][r] + bb;
        }
    }
}

extern "C" void kernel_launch(void* const* d_in, const int* in_sizes, int n_in,
                              void* d_out, int out_size, void* d_ws, size_t ws_size,
                              hipStream_t stream) {
    const float* nh  = (const float*)d_in[0];
    const float* eh  = (const float*)d_in[1];
    const int*   ei  = (const int*)d_in[2];   // [2, E] int32
    const float* W1  = (const float*)d_in[3];
    const float* b1  = (const float*)d_in[4];
    const float* W2  = (const float*)d_in[5];
    const float* b2  = (const float*)d_in[6];
    float* out = (float*)d_out;

    const int* src = ei;
    const int* dst = ei + N_EDGES;

    float* deg = (float*)d_ws;            // [N_NODES]
    float* dis = deg + N_NODES;           // [N_NODES]

    // zero deg and the agg region (first part of d_out used as agg scratch)
    hipMemsetAsync(deg, 0, N_NODES * sizeof(float), stream);
    hipMemsetAsync(out, 0, (size_t)N_NODES * D * sizeof(float), stream);

    deg_kernel<<<(N_EDGES + 255) / 256, 256, 0, stream>>>(dst, deg);
    dis_kernel<<<(N_NODES + 255) / 256, 256, 0, stream>>>(deg, dis);

    long long sthreads = (long long)N_EDGES * 32;
    scatter_kernel<<<(unsigned)((sthreads + 255) / 256), 256, 0, stream>>>(
        nh, src, dst, dis, out);

    size_t shmem = (size_t)(2 * 128 * WPITCH + 256 + MLP_WAVES * 16 * WPITCH)
                   * sizeof(float);
    mlp_kernel<<<(N_TILES + MLP_WAVES - 1) / MLP_WAVES, 32 * MLP_WAVES, shmem,
                 stream>>>(dis, W1, b1, W2, b2, out);

    // eh pass-through into second output slot
    hipMemcpyAsync(out + (size_t)N_NODES * D, eh,
                   (size_t)N_EDGES * 16 * sizeof(float),
                   hipMemcpyDeviceToDevice, stream);
}